// Self_L2_53927609369069
// MI455X (gfx1250) — compile-verified
//
#include <hip/hip_runtime.h>
#include <math.h>

typedef __attribute__((ext_vector_type(2))) float v2f;
typedef __attribute__((ext_vector_type(8))) float v8f;

#define LOG_EPS (-9.210340371976182f) /* log(1e-4) */

// W[c][k] = T_k[c] = 1.0 if c==k else 1e-4 (c,k < 3); zero padding elsewhere.
__device__ __forceinline__ float Wmat(int K, int n) {
    if (K >= 3 || n >= 3) return 0.0f;
    return (K == n) ? 1.0f : 1.0e-4f;
}

// One wave = 16 rows. v_{i,k} = dot(D[i,:], T_k) via V_WMMA_F32_16X16X4_F32.
// Per block: 8 waves -> 128 rows -> 3 per-class |v| partial sums + 3 class counts.
__global__ __launch_bounds__(256) void Self_L2_partial_kernel(
    const float* __restrict__ s, const int* __restrict__ t,
    float* __restrict__ wsS, int* __restrict__ wsC, int B)
{
    __shared__ float lsp[8][3][2];
    __shared__ int   lcnt[3];

    const int tid  = threadIdx.x;
    const int w    = tid >> 5;        // wave in block (0..7)
    const int ln   = tid & 31;        // lane in wave
    const int n    = ln & 15;         // column / M-row-within-16
    const int half = ln >> 4;         // lane half selects K pair
    const int rowBase = blockIdx.x * 128 + w * 16;
    const int m    = rowBase + n;     // this lane's D row

    if (tid < 3) lcnt[tid] = 0;
    __syncthreads();

    // ---- per-row D = logT - log_softmax(s) (computed in both lane halves) ----
    float D0 = 0.0f, D1 = 0.0f, D2 = 0.0f;
    int   cls = 0;
    const bool valid = (m < B);
    if (valid) {
        const float s0 = s[3 * m + 0];
        const float s1 = s[3 * m + 1];
        const float s2 = s[3 * m + 2];
        cls = t[m];
        const float mx  = fmaxf(s0, fmaxf(s1, s2));
        const float lse = mx + logf(expf(s0 - mx) + expf(s1 - mx) + expf(s2 - mx));
        const float lt0 = (cls == 0) ? 0.0f : LOG_EPS;
        const float lt1 = (cls == 1) ? 0.0f : LOG_EPS;
        const float lt2 = (cls == 2) ? 0.0f : LOG_EPS;
        D0 = lt0 - (s0 - lse);
        D1 = lt1 - (s1 - lse);
        D2 = lt2 - (s2 - lse);
    }

    // ---- A operand: 16x4 f32, lane L<16 holds K={0,1}, L>=16 holds K={2,3} ----
    v2f a;
    a.x = (half == 0) ? D0 : D2;
    a.y = (half == 0) ? D1 : 0.0f;   // K=3 column is zero padding

    // ---- B operand: 4x16 f32, VGPR0: K={0|2}, VGPR1: K={1|3}, lane = column ----
    v2f b;
    b.x = (half == 0) ? Wmat(0, n) : Wmat(2, n);
    b.y = (half == 0) ? Wmat(1, n) : Wmat(3, n);

    // ---- WMMA: D16x16 = A(16x4) x B(4x16); EXEC is all-ones here ----
    v8f c = {};
    c = __builtin_amdgcn_wmma_f32_16x16x4_f32(
        /*neg_a=*/false, a, /*neg_b=*/false, b,
        /*c_mod=*/(short)0, c, /*reuse_a=*/false, /*reuse_b=*/false);

    // lane holds rows rowBase + v + 8*half, column n == class (valid if n<3)
    float p = 0.0f;
#pragma unroll
    for (int v = 0; v < 8; ++v) p += fabsf(c[v]);

    if (n < 3) lsp[w][n][half] = p;
    if (half == 0 && valid) atomicAdd(&lcnt[cls], 1);
    __syncthreads();

    // fixed-order (deterministic) block reduction, 3 threads
    if (tid < 3) {
        float bs = 0.0f;
        for (int ww = 0; ww < 8; ++ww)
            for (int h = 0; h < 2; ++h)
                bs += lsp[ww][tid][h];
        wsS[blockIdx.x * 3 + tid] = bs;
        wsC[blockIdx.x * 3 + tid] = lcnt[tid];
    }
}

__global__ void Self_L2_finalize_kernel(
    const float* __restrict__ wsS, const int* __restrict__ wsC,
    int nblk, float* __restrict__ out, float invB2)
{
    if (threadIdx.x == 0 && blockIdx.x == 0) {
        float total = 0.0f;
        for (int k = 0; k < 3; ++k) {
            float S = 0.0f;
            int   C = 0;
            for (int bIdx = 0; bIdx < nblk; ++bIdx) {
                S += wsS[bIdx * 3 + k];
                C += wsC[bIdx * 3 + k];
            }
            total += (float)C * S;
        }
        out[0] = 1.0e-4f + total * invB2;
    }
}

extern "C" void kernel_launch(void* const* d_in, const int* in_sizes, int n_in,
                              void* d_out, int out_size, void* d_ws, size_t ws_size,
                              hipStream_t stream) {
    (void)n_in; (void)out_size; (void)ws_size;
    const float* s = (const float*)d_in[0];
    const int*   t = (const int*)d_in[1];
    const int B    = in_sizes[1];          // number of rows (t element count)
    const int nblk = (B + 127) / 128;      // 128 rows per block (8 waves x 16)

    float* wsS = (float*)d_ws;
    int*   wsC = (int*)((char*)d_ws + (size_t)nblk * 3 * sizeof(float));

    Self_L2_partial_kernel<<<nblk, 256, 0, stream>>>(s, t, wsS, wsC, B);
    Self_L2_finalize_kernel<<<1, 32, 0, stream>>>(
        wsS, wsC, nblk, (float*)d_out, 1.0f / ((float)B * (float)B));
}